// EmbGCN_36816459661693
// MI455X (gfx1250) — compile-verified
//
#include <hip/hip_runtime.h>
#include <math.h>

typedef __attribute__((ext_vector_type(2))) float v2f;
typedef __attribute__((ext_vector_type(8))) float v8f;

__device__ __forceinline__ float eluf(float v) { return v > 0.0f ? v : expm1f(v); }

// ---------------------------------------------------------------------------
// Small elementwise / edge kernels
// ---------------------------------------------------------------------------
__global__ __launch_bounds__(256) void k_fill1(float* __restrict__ p, int n) {
    int i = blockIdx.x * blockDim.x + threadIdx.x;
    if (i < n) p[i] = 1.0f;   // self-loop weight 1 seeds the degree
}

__global__ __launch_bounds__(256) void k_deg_acc(const int* __restrict__ dst,
                                                 const float* __restrict__ ew,
                                                 float* __restrict__ deg, int e) {
    int i = blockIdx.x * blockDim.x + threadIdx.x;
    if (i < e) atomicAdd(&deg[dst[i]], ew[i]);
}

__global__ __launch_bounds__(256) void k_dinv(const float* __restrict__ deg,
                                              float* __restrict__ dinv, int n) {
    int i = blockIdx.x * blockDim.x + threadIdx.x;
    if (i < n) dinv[i] = rsqrtf(deg[i]);   // deg >= 1 always (self loop)
}

__global__ __launch_bounds__(256) void k_norm(const int* __restrict__ src,
                                              const int* __restrict__ dst,
                                              const float* __restrict__ ew,
                                              const float* __restrict__ dinv,
                                              float* __restrict__ nrm, int e) {
    int i = blockIdx.x * blockDim.x + threadIdx.x;
    if (i < e) nrm[i] = dinv[src[i]] * ew[i] * dinv[dst[i]];
}

__global__ __launch_bounds__(256) void k_elu_bias(float* __restrict__ x,
                                                  const float* __restrict__ b, int total) {
    int i = blockIdx.x * blockDim.x + threadIdx.x;
    if (i < total) x[i] = eluf(x[i] + b[i & 63]);
}

// ---------------------------------------------------------------------------
// WMMA GEMM: Y[nrows,64] = X[rowidx,64] @ W[64,64]  (fp32, V_WMMA_F32_16X16X4_F32)
// MODE 0: Y = XW, AGG = XW * dinv[row]^2       (GCN conv pre-aggregation)
// MODE 1: Y = ELU(XW + bias)                   (dense linear layer)
// One wave -> 16 rows x 64 cols (4 tiles), K=64 in 16 steps of 4.
// ---------------------------------------------------------------------------
template <int MODE>
__global__ __launch_bounds__(128)
void k_gemm64(const float* __restrict__ X, const float* __restrict__ W,
              const float* __restrict__ bias, const float* __restrict__ dinv,
              const int* __restrict__ rowidx,
              float* __restrict__ Y, float* __restrict__ AGG, int nrows) {
    // Stage W transposed into LDS: lw[n*68 + k] = W[k][n]; stride 68 kills bank conflicts.
    __shared__ float lw[64 * 68];
    for (int i = threadIdx.x; i < 64 * 64; i += 128) {
        int k = i >> 6, n = i & 63;
        lw[n * 68 + k] = W[i];
    }
    __syncthreads();

    const int lane = threadIdx.x & 31;
    const int wave = threadIdx.x >> 5;
    const int rowBase = (blockIdx.x * 4 + wave) * 16;
    if (rowBase >= nrows) return;          // wave-uniform: EXEC stays all-ones for WMMA

    const int nl = lane & 15;              // row (A) / col (B) within tile
    const int hh = lane >> 4;              // 0: K pair {0,1}; 1: K pair {2,3}
    const int kh = hh * 2;

    int m = rowBase + nl;
    if (m >= nrows) m = nrows - 1;         // N % 16 == 0 here, but stay safe
    const long arow = rowidx ? (long)rowidx[m] : (long)m;
    const v2f* __restrict__ ap = (const v2f*)(X + arow * 64);

    v8f acc[4] = {};
#pragma unroll
    for (int ks = 0; ks < 16; ++ks) {
        // A fragment (16x4): lane nl holds X[row][ks*4+kh .. +1]
        v2f a = ap[ks * 2 + hh];
#pragma unroll
        for (int t = 0; t < 4; ++t) {
            // B fragment (4x16): lane nl holds W[ks*4+kh][t*16+nl], W[ks*4+kh+1][..]
            v2f b = *(const v2f*)&lw[(t * 16 + nl) * 68 + ks * 4 + kh];
            acc[t] = __builtin_amdgcn_wmma_f32_16x16x4_f32(
                false, a, false, b, (short)0, acc[t], false, false);
        }
    }

    // C/D layout: VGPR r -> row (rowBase + r + 8*hh), col = t*16 + nl
    const int rAdd = hh * 8;
#pragma unroll
    for (int r = 0; r < 8; ++r) {
        const int row = rowBase + r + rAdd;
        if (row < nrows) {
            if (MODE == 0) {
                const float di = dinv[row];
                const float d2 = di * di;
#pragma unroll
                for (int t = 0; t < 4; ++t) {
                    const int col = t * 16 + nl;
                    const float v = acc[t][r];
                    Y[(long)row * 64 + col] = v;
                    AGG[(long)row * 64 + col] = v * d2;   // self-loop term
                }
            } else {
#pragma unroll
                for (int t = 0; t < 4; ++t) {
                    const int col = t * 16 + nl;
                    Y[(long)row * 64 + col] = eluf(acc[t][r] + bias[col]);
                }
            }
        }
    }
}

// ---------------------------------------------------------------------------
// Edge scatter: agg[dst] += xw[src] * norm.  16 threads/edge, float4 per thread.
// xw (25.6 MB) and agg are L2-resident; atomics resolve in L2.
// ---------------------------------------------------------------------------
__global__ __launch_bounds__(256)
void k_scatter(const float* __restrict__ xw, const float* __restrict__ nrm,
               const int* __restrict__ src, const int* __restrict__ dst,
               float* __restrict__ agg, int e) {
    int tid = blockIdx.x * blockDim.x + threadIdx.x;
    int ei = tid >> 4;
    if (ei >= e) return;
    int f = (tid & 15) << 2;
    int s = src[ei], d = dst[ei];
    float nm = nrm[ei];
    const float4 v = *(const float4*)(xw + (long)s * 64 + f);
    float* o = agg + (long)d * 64 + f;
    atomicAdd(o + 0, v.x * nm);
    atomicAdd(o + 1, v.y * nm);
    atomicAdd(o + 2, v.z * nm);
    atomicAdd(o + 3, v.w * nm);
}

// ---------------------------------------------------------------------------
// Fused tail: x = x1 + ELU(agg2 + b2); logits = x @ Wl + bl; log_softmax.
// One thread per node; Wl/bl/b2 broadcast from LDS (conflict-free).
// ---------------------------------------------------------------------------
__global__ __launch_bounds__(256)
void k_head(const float* __restrict__ x1, const float* __restrict__ agg2,
            const float* __restrict__ b2, const float* __restrict__ Wl,
            const float* __restrict__ bl, float* __restrict__ out, int n) {
    __shared__ float sW[64 * 16];
    __shared__ float sb2[64];
    __shared__ float sbl[16];
    for (int i = threadIdx.x; i < 64 * 16; i += 256) sW[i] = Wl[i];
    if (threadIdx.x < 64) sb2[threadIdx.x] = b2[threadIdx.x];
    if (threadIdx.x < 16) sbl[threadIdx.x] = bl[threadIdx.x];
    __syncthreads();

    int node = blockIdx.x * blockDim.x + threadIdx.x;
    if (node >= n) return;

    float x[64];
    const float4* a4 = (const float4*)(agg2 + (long)node * 64);
    const float4* p4 = (const float4*)(x1 + (long)node * 64);
#pragma unroll
    for (int q = 0; q < 16; ++q) {
        float4 av = a4[q], pv = p4[q];
        int k = q * 4;
        x[k + 0] = pv.x + eluf(av.x + sb2[k + 0]);
        x[k + 1] = pv.y + eluf(av.y + sb2[k + 1]);
        x[k + 2] = pv.z + eluf(av.z + sb2[k + 2]);
        x[k + 3] = pv.w + eluf(av.w + sb2[k + 3]);
    }

    float logit[16];
#pragma unroll
    for (int c = 0; c < 16; ++c) logit[c] = sbl[c];
#pragma unroll
    for (int k = 0; k < 64; ++k) {
        const float xv = x[k];
#pragma unroll
        for (int c = 0; c < 16; ++c) logit[c] = fmaf(xv, sW[k * 16 + c], logit[c]);
    }

    float mx = logit[0];
#pragma unroll
    for (int c = 1; c < 16; ++c) mx = fmaxf(mx, logit[c]);
    float se = 0.0f;
#pragma unroll
    for (int c = 0; c < 16; ++c) se += expf(logit[c] - mx);
    const float lse = mx + logf(se);
    float* o = out + (long)node * 16;
#pragma unroll
    for (int c = 0; c < 16; ++c) o[c] = logit[c] - lse;
}

// ---------------------------------------------------------------------------
extern "C" void kernel_launch(void* const* d_in, const int* in_sizes, int n_in,
                              void* d_out, int out_size, void* d_ws, size_t ws_size,
                              hipStream_t stream) {
    const int*   node_index = (const int*)d_in[0];
    const int*   edge       = (const int*)d_in[1];
    const float* ew         = (const float*)d_in[2];
    const float* emb        = (const float*)d_in[3];
    const float* W1 = (const float*)d_in[4];
    const float* b1 = (const float*)d_in[5];
    const float* W2 = (const float*)d_in[6];
    const float* b2 = (const float*)d_in[7];
    const float* Wf = (const float*)d_in[8];
    const float* bf = (const float*)d_in[9];
    const float* Wl = (const float*)d_in[10];
    const float* bl = (const float*)d_in[11];

    const int N = in_sizes[0];
    const int E = in_sizes[1] / 2;
    const int* src = edge;          // edge_index[0]
    const int* dst = edge + E;      // edge_index[1]
    float* out = (float*)d_out;

    // workspace carve-up (all fp32, 16B aligned chunks)
    float* ws = (float*)d_ws;
    size_t off = 0;
    auto carve = [&](size_t cnt) { float* p = ws + off; off += (cnt + 3) & ~(size_t)3; return p; };
    float* deg  = carve((size_t)N);
    float* dinv = carve((size_t)N);
    float* nrm  = carve((size_t)E);
    float* xw   = carve((size_t)N * 64);   // reused for both convs
    float* agg1 = carve((size_t)N * 64);   // becomes x1 in place
    float* xf   = carve((size_t)N * 64);
    float* agg2 = carve((size_t)N * 64);

    const int B = 256;
    const unsigned gN  = (unsigned)((N + B - 1) / B);
    const unsigned gE  = (unsigned)((E + B - 1) / B);
    const unsigned gNE = (unsigned)(((size_t)N * 64 + B - 1) / B);
    const unsigned gSc = (unsigned)(((size_t)E * 16 + B - 1) / B);
    const unsigned gG  = (unsigned)((N + 63) / 64);

    // degrees -> dinv -> per-edge norm (shared by both convs)
    k_fill1  <<<gN, B, 0, stream>>>(deg, N);
    k_deg_acc<<<gE, B, 0, stream>>>(dst, ew, deg, E);
    k_dinv   <<<gN, B, 0, stream>>>(deg, dinv, N);
    k_norm   <<<gE, B, 0, stream>>>(src, dst, ew, dinv, nrm, E);

    // conv1: xw1 = emb[node_index] @ W1 ; agg1 seeded with self-loop term
    k_gemm64<0><<<gG, 128, 0, stream>>>(emb, W1, nullptr, dinv, node_index, xw, agg1, N);
    k_scatter<<<gSc, B, 0, stream>>>(xw, nrm, src, dst, agg1, E);
    k_elu_bias<<<gNE, B, 0, stream>>>(agg1, b1, N * 64);          // agg1 -> x1

    // first_lin: xf = ELU(emb[node_index] @ Wf + bf)
    k_gemm64<1><<<gG, 128, 0, stream>>>(emb, Wf, bf, nullptr, node_index, xf, nullptr, N);

    // conv2: xw2 = xf @ W2 ; agg2 seeded, then scatter
    k_gemm64<0><<<gG, 128, 0, stream>>>(xf, W2, nullptr, dinv, nullptr, xw, agg2, N);
    k_scatter<<<gSc, B, 0, stream>>>(xw, nrm, src, dst, agg2, E);

    // fused: skip + bias + ELU + classifier + log_softmax
    k_head<<<gN, B, 0, stream>>>(agg1, agg2, b2, Wl, bl, out, N);
}